// find_idx_40707700032017
// MI455X (gfx1250) — compile-verified
//
#include <hip/hip_runtime.h>
#include <math.h>

#define H_DIM 512
#define W_DIM 512
#define HOLE_DIM 128
#define NU (HOLE_DIM * HOLE_DIM)        // 16384 unfilled (queries)
#define NF (H_DIM * W_DIM - NU)         // 245760 filled (references)
#define MT 128                          // queries per block (8 waves x 16)
#define TF 1024                         // filled points per LDS tile

static_assert(NF % TF == 0, "staging loop assumes exact tiling");
static_assert(NU % MT == 0, "grid assumes exact tiling");

typedef __attribute__((ext_vector_type(2))) float v2f;
typedef __attribute__((ext_vector_type(4))) float v4f;
typedef __attribute__((ext_vector_type(8))) float v8f;
typedef unsigned long long u64;

// d_out layout (4-byte elements), concatenated in reference return order:
//   [0,          NU*3)              idx      (int32)
//   [NU*3,       NU*6)              dist     (float32)
//   [NU*6,       NU*6+NF*2)         filled   (int32 row,col)
//   [NU*6+NF*2,  NU*6+NF*2+NU*2)    unfilled (int32 row,col)

// Branchless sorted-3 insert on u64 keys = (f32 bits of d^2 << 32) | idx.
// d^2 >= 0 so float ordering == unsigned ordering of the bits; lowered by
// clang to v_min_u64/v_max_u64 chains -> exact (d, idx) top_k tie-break.
__device__ __forceinline__ void top3_insert(u64 k, u64 (&b)[3]) {
  u64 t = (k < b[2]) ? k : b[2];
  bool l1 = (t < b[1]);
  u64 m = l1 ? t : b[1];
  b[2] = l1 ? b[1] : t;
  bool l0 = (m < b[0]);
  b[1] = l0 ? b[0] : m;
  b[0] = l0 ? m : b[0];
}

__device__ __forceinline__ u64 shflx64(u64 v, int m) {
  int lo = __shfl_xor((int)(unsigned)v, m, 32);
  int hi = __shfl_xor((int)(unsigned)(v >> 32), m, 32);
  return ((u64)(unsigned)hi << 32) | (unsigned)lo;
}

// Kernel 1: per-row unfilled counts + exclusive scan -> row base offsets.
__global__ __launch_bounds__(512) void row_scan_kernel(const float* __restrict__ mask,
                                                       int* __restrict__ rowUoff,
                                                       int* __restrict__ rowFoff) {
  __shared__ int su[H_DIM];
  const int r = threadIdx.x;
  int cnt = 0;
  for (int c = 0; c < W_DIM; ++c)
    cnt += (mask[r * W_DIM + c] == 0.0f) ? 1 : 0;
  su[r] = cnt;
  __syncthreads();
  for (int off = 1; off < H_DIM; off <<= 1) {
    int v = (r >= off) ? su[r - off] : 0;
    __syncthreads();
    su[r] += v;
    __syncthreads();
  }
  const int excl = su[r] - cnt;          // unfilled before row r
  rowUoff[r] = excl;
  rowFoff[r] = r * W_DIM - excl;         // filled before row r
}

// Kernel 2: row-major compaction of coordinates (matches jnp.nonzero order).
__global__ __launch_bounds__(512) void emit_kernel(const float* __restrict__ mask,
                                                   const int* __restrict__ rowUoff,
                                                   const int* __restrict__ rowFoff,
                                                   int* __restrict__ out) {
  __shared__ int sc[W_DIM];
  const int r = blockIdx.x;
  const int c = threadIdx.x;
  const int u = (mask[r * W_DIM + c] == 0.0f) ? 1 : 0;
  sc[c] = u;
  __syncthreads();
  for (int off = 1; off < W_DIM; off <<= 1) {
    int v = (c >= off) ? sc[c - off] : 0;
    __syncthreads();
    sc[c] += v;
    __syncthreads();
  }
  const int uex = sc[c] - u;             // unfilled before col c in row
  const int fex = c - uex;               // filled before col c in row
  int* __restrict__ filled   = out + (size_t)NU * 6;
  int* __restrict__ unfilled = out + (size_t)NU * 6 + (size_t)NF * 2;
  if (u) {
    const int p = rowUoff[r] + uex;
    unfilled[p * 2 + 0] = r;
    unfilled[p * 2 + 1] = c;
  } else {
    const int p = rowFoff[r] + fex;
    filled[p * 2 + 0] = r;
    filled[p * 2 + 1] = c;
  }
}

// One insert block for a 16x16 WMMA d^2 tile; returns updated threshold w.
__device__ __forceinline__ float tile_update(const v8f& dd, int n, float w,
                                             u64 (&b)[8][3]) {
  // Wave-coherent early-out: any slot possibly improved?
  float dmin = fminf(fminf(fminf(dd[0], dd[1]), fminf(dd[2], dd[3])),
                     fminf(fminf(dd[4], dd[5]), fminf(dd[6], dd[7])));
  if (dmin <= w) {                       // one v_cmpx + s_cbranch_execz
#pragma unroll
    for (int s = 0; s < 8; ++s) {
      u64 k = ((u64)__float_as_uint(dd[s]) << 32) | (unsigned)n;
      top3_insert(k, b[s]);              // v_min_u64/v_max_u64 chain
    }
    float w0 = fmaxf(__uint_as_float((unsigned)(b[0][2] >> 32)),
                     __uint_as_float((unsigned)(b[1][2] >> 32)));
    float w1 = fmaxf(__uint_as_float((unsigned)(b[2][2] >> 32)),
                     __uint_as_float((unsigned)(b[3][2] >> 32)));
    float w2 = fmaxf(__uint_as_float((unsigned)(b[4][2] >> 32)),
                     __uint_as_float((unsigned)(b[5][2] >> 32)));
    float w3 = fmaxf(__uint_as_float((unsigned)(b[6][2] >> 32)),
                     __uint_as_float((unsigned)(b[7][2] >> 32)));
    w = fmaxf(fmaxf(w0, w1), fmaxf(w2, w3));
  }
  return w;
}

// Kernel 3: brute-force 3-NN. One V_WMMA_F32_16X16X4_F32 produces an exact
// 16x16 tile of d^2: A-row=(-2ur,-2uc,|u|^2,1), B-col=(fr,fc,1,|f|^2).
__global__ __launch_bounds__(256) void knn_wmma_kernel(int* __restrict__ out) {
  __shared__ v4f ftile[TF + 32];         // (fr, fc, 1, |f|^2); +32 pipeline pad
  const int tid  = threadIdx.x;
  const int lane = tid & 31;
  const int wave = tid >> 5;
  const int half = lane >> 4;            // 0: K=0..1 / rows 0..7 ; 1: K=2..3 / rows 8..15
  const int l16  = lane & 15;
  const int qb   = blockIdx.x * MT + wave * 16;   // 16 queries per wave

  int*   __restrict__ idx_out  = out;
  float* __restrict__ dist_out = (float*)(out + (size_t)NU * 3);
  const int* __restrict__ filled   = out + (size_t)NU * 6;
  const int* __restrict__ unfilled = out + (size_t)NU * 6 + (size_t)NF * 2;

  // A fragment (16x4 f32 layout: lanes 0-15 hold K0,K1 of row M=lane;
  // lanes 16-31 hold K2,K3 of row M=lane-16).
  const int   qm = qb + l16;
  const float ur = (float)unfilled[qm * 2 + 0];
  const float uc = (float)unfilled[qm * 2 + 1];
  v2f afrag;
  if (half == 0) { afrag[0] = -2.0f * ur; afrag[1] = -2.0f * uc; }
  else           { afrag[0] = ur * ur + uc * uc; afrag[1] = 1.0f; }

  // Sorted top-3 keys per slot; finite sentinel (NaN bits would break the guard).
  const u64 SENT = ((u64)__float_as_uint(3.0e38f) << 32) | 0x7FFFFFFFull;
  u64 b[8][3];
#pragma unroll
  for (int s = 0; s < 8; ++s) { b[s][0] = SENT; b[s][1] = SENT; b[s][2] = SENT; }
  float w = 3.0e38f;                     // cached max over slots of 3rd-best d^2

  const v2f* bb = ((const v2f*)ftile) + (l16 * 2 + half);   // lane's B-frag column

  for (int nb = 0; nb < NF; nb += TF) {
    __syncthreads();
    // Stage TF filled points into LDS as (fr, fc, 1, |f|^2)  (ds_store_b128).
#pragma unroll
    for (int t = tid; t < TF; t += 256) {
      const int p = nb + t;              // always < NF (NF % TF == 0)
      const float fr = (float)filled[p * 2 + 0];
      const float fc = (float)filled[p * 2 + 1];
      v4f v; v[0] = fr; v[1] = fc; v[2] = 1.0f; v[3] = fr * fr + fc * fc;
      ftile[t] = v;
    }
    if (nb + TF < NF)                                 // gfx1250 global_prefetch_b8
      __builtin_prefetch(&filled[(size_t)(nb + TF) * 2 + (size_t)tid * 8], 0, 3);
    __syncthreads();

    // Two B fragments in flight -> loads stay 1 iteration ahead of use.
    v2f frag0 = bb[0];                   // points j..j+15
    v2f frag1 = bb[32];                  // points j+16..j+31
    for (int j = 0; j < TF; j += 32) {
      v2f fn0 = bb[(j + 32) * 2];        // pad makes tail reads safe
      v2f fn1 = bb[(j + 48) * 2];
      v8f cz0 = {};
      v8f dd0 = __builtin_amdgcn_wmma_f32_16x16x4_f32(
          false, afrag, false, frag0, (short)0, cz0, false, false);
      w = tile_update(dd0, nb + j + l16, w, b);
      v8f cz1 = {};
      v8f dd1 = __builtin_amdgcn_wmma_f32_16x16x4_f32(
          false, afrag, false, frag1, (short)0, cz1, false, false);
      w = tile_update(dd1, nb + j + 16 + l16, w, b);
      frag0 = fn0;
      frag1 = fn1;
    }
  }

  // Merge top-3 across the 16 lanes of each half (butterfly on u64 keys).
#pragma unroll
  for (int s = 0; s < 8; ++s) {
#pragma unroll
    for (int msk = 1; msk <= 8; msk <<= 1) {
      u64 o0 = shflx64(b[s][0], msk);
      u64 o1 = shflx64(b[s][1], msk);
      u64 o2 = shflx64(b[s][2], msk);
      top3_insert(o0, b[s]);
      top3_insert(o1, b[s]);
      top3_insert(o2, b[s]);
    }
  }

  if (l16 == 0) {
    const int base = qb + half * 8;
#pragma unroll
    for (int s = 0; s < 8; ++s) {
      const int q = base + s;
#pragma unroll
      for (int k = 0; k < 3; ++k) {
        idx_out[q * 3 + k] = (int)(unsigned)b[s][k];
        const float d2 = __uint_as_float((unsigned)(b[s][k] >> 32));
        dist_out[q * 3 + k] = sqrtf(d2 > 0.0f ? d2 : 0.0f);
      }
    }
  }
}

extern "C" void kernel_launch(void* const* d_in, const int* in_sizes, int n_in,
                              void* d_out, int out_size, void* d_ws, size_t ws_size,
                              hipStream_t stream) {
  (void)in_sizes; (void)n_in; (void)out_size; (void)ws_size;
  const float* mask = (const float*)d_in[0];
  int* out = (int*)d_out;
  int* rowUoff = (int*)d_ws;           // 512 ints
  int* rowFoff = rowUoff + H_DIM;      // 512 ints

  row_scan_kernel<<<1, H_DIM, 0, stream>>>(mask, rowUoff, rowFoff);
  emit_kernel<<<H_DIM, W_DIM, 0, stream>>>(mask, rowUoff, rowFoff, out);
  knn_wmma_kernel<<<NU / MT, 256, 0, stream>>>(out);
}